// CEProbs_34102040330686
// MI455X (gfx1250) — compile-verified
//
#include <hip/hip_runtime.h>
#include <stdint.h>

// CE-with-soft-labels reduction for MI455X (gfx1250).
//   out = mean_i [ (max_i + log(sum_j exp(p_ij - max_i))) * sum_j t_ij - sum_j t_ij*p_ij ]
// Bandwidth-bound: stream P and T exactly once from HBM via async LDS staging,
// do the 2-pass logsumexp out of LDS, fold row sums into V_WMMA_F32_16X16X4_F32.

#define C_DIM 1000
#define C_PAD 1024
#define WAVES_PER_BLOCK 4
#define BLOCK_THREADS (WAVES_PER_BLOCK * 32)

typedef __attribute__((ext_vector_type(2))) float v2f;
typedef __attribute__((ext_vector_type(8))) float v8f;

__device__ __forceinline__ uint32_t lds_addr_of(const void* p) {
  // Generic (flat) shared-space pointer: low 32 bits are the LDS byte offset.
  return (uint32_t)(uintptr_t)p;
}

__global__ __launch_bounds__(BLOCK_THREADS) void CEProbs_34102040330686_kernel(
    const float* __restrict__ predicted, const float* __restrict__ target,
    float* __restrict__ out, int n_rows, float inv_n) {
  __shared__ float pbuf[WAVES_PER_BLOCK][C_PAD];
  __shared__ float tbuf[WAVES_PER_BLOCK][C_PAD];
  __shared__ float block_acc;

  const int lane = threadIdx.x & 31;
  const int wave = threadIdx.x >> 5;
  if (threadIdx.x == 0) block_acc = 0.0f;
  __syncthreads();

  const int gwave = blockIdx.x * WAVES_PER_BLOCK + wave;
  const int nwaves = gridDim.x * WAVES_PER_BLOCK;

  float local_acc = 0.0f;

  for (int row = gwave; row < n_rows; row += nwaves) {
    const float* prow = predicted + (size_t)row * C_DIM;
    const float* trow = target + (size_t)row * C_DIM;

    // ---- stage row into LDS with CDNA5 async tensor-path (ASYNCcnt) ----
    // 8 issues x 32 lanes x b128 = 1024 floats per array; tail lanes clamp
    // their global address into the row (data discarded, lands in pad area).
    #pragma unroll
    for (int c = 0; c < 8; ++c) {
      const int e = c * 128 + lane * 4;            // first element this lane copies
      const int ge = (e + 4 <= C_DIM) ? e : 0;     // clamp OOB tail reads
      uint64_t gp = (uint64_t)(uintptr_t)(prow + ge);
      uint64_t gt = (uint64_t)(uintptr_t)(trow + ge);
      uint32_t lp = lds_addr_of(&pbuf[wave][e]);
      uint32_t lt = lds_addr_of(&tbuf[wave][e]);
      asm volatile("global_load_async_to_lds_b128 %0, %1, off"
                   :: "v"(lp), "v"(gp) : "memory");
      asm volatile("global_load_async_to_lds_b128 %0, %1, off"
                   :: "v"(lt), "v"(gt) : "memory");
    }
    asm volatile("s_wait_asynccnt 0x0" ::: "memory");

    // Repair the pad region [C_DIM, C_PAD): exp() -> 0, t -> 0.
    if (lane < C_PAD - C_DIM) {
      pbuf[wave][C_DIM + lane] = -1.0e30f;
      tbuf[wave][C_DIM + lane] = 0.0f;
    }

    // ---- pass 1: row max (VALU + wave32 shuffle tree) ----
    float m = -1.0e30f;
    #pragma unroll
    for (int i = 0; i < C_PAD / 32; ++i)
      m = fmaxf(m, pbuf[wave][lane + 32 * i]);
    #pragma unroll
    for (int off = 16; off > 0; off >>= 1)
      m = fmaxf(m, __shfl_xor(m, off, 32));

    // ---- pass 2: WMMA-folded row sums of exp(p-m), t, t*p ----
    // A = 16x4 f32 tile (2 elems/lane), B = ones(4x16) => D accumulates row
    // sums replicated across columns. Only the total matters, so the exact
    // lane->(M,K) permutation is irrelevant.
    v8f accE = {};
    v8f accT = {};
    v8f accTP = {};
    const v2f ones = {1.0f, 1.0f};
    #pragma unroll
    for (int base = 0; base < C_PAD; base += 64) {
      const float2 p2 = *(const float2*)&pbuf[wave][base + 2 * lane];
      const float2 t2 = *(const float2*)&tbuf[wave][base + 2 * lane];
      v2f e  = {__expf(p2.x - m), __expf(p2.y - m)};
      v2f t  = {t2.x, t2.y};
      v2f tp = {t2.x * p2.x, t2.y * p2.y};
      accE  = __builtin_amdgcn_wmma_f32_16x16x4_f32(false, e,  false, ones,
                                                    (short)0, accE,  false, false);
      accT  = __builtin_amdgcn_wmma_f32_16x16x4_f32(false, t,  false, ones,
                                                    (short)0, accT,  false, false);
      accTP = __builtin_amdgcn_wmma_f32_16x16x4_f32(false, tp, false, ones,
                                                    (short)0, accTP, false, false);
    }
    // Lanes 0-15 hold rows 0-7 of D, lanes 16-31 hold rows 8-15 (col-replicated).
    float sE = 0.0f, sT = 0.0f, sTP = 0.0f;
    #pragma unroll
    for (int i = 0; i < 8; ++i) {
      sE += accE[i];
      sT += accT[i];
      sTP += accTP[i];
    }
    sE  += __shfl_xor(sE, 16, 32);
    sT  += __shfl_xor(sT, 16, 32);
    sTP += __shfl_xor(sTP, 16, 32);

    if (lane == 0) {
      const float lse = m + __logf(sE);
      local_acc += lse * sT - sTP;
    }
  }

  if (lane == 0) atomicAdd(&block_acc, local_acc);
  __syncthreads();
  if (threadIdx.x == 0) atomicAdd(out, block_acc * inv_n);
}

extern "C" void kernel_launch(void* const* d_in, const int* in_sizes, int n_in,
                              void* d_out, int out_size, void* d_ws, size_t ws_size,
                              hipStream_t stream) {
  const float* predicted = (const float*)d_in[0];
  const float* target = (const float*)d_in[1];
  float* out = (float*)d_out;
  const int n_rows = in_sizes[0] / C_DIM;

  hipMemsetAsync(out, 0, sizeof(float), stream);

  const int blocks = 4096;  // 16384 wave32s -> 8 rows per wave at N=131072
  CEProbs_34102040330686_kernel<<<blocks, BLOCK_THREADS, 0, stream>>>(
      predicted, target, out, n_rows, 1.0f / (float)n_rows);
}